// SampleGroupMoudle_51144470561304
// MI455X (gfx1250) — compile-verified
//
#include <hip/hip_runtime.h>

// Farthest point sampling for B=16, N=131072, npoint=4096.
//   - 8 workgroups per batch (128 WGs total), 512 threads each.
//   - xyz tile staged into LDS via CDNA5 global_load_async_to_lds_b128
//     (ASYNCcnt / s_wait_asynccnt), stored as float4(x,y,z,|p|^2) = 256KB LDS.
//   - running min-distance kept in VGPRs (32 per thread).
//   - cross-WG argmax via packed u64 global_atomic_max + monotonic counter
//     barrier in d_ws (triple-buffered slots, reset each call by init kernel).

#define WPB      8                       // workgroups per batch
#define THREADS  512
#define WAVES    (THREADS / 32)

constexpr int    kN            = 131072;
constexpr int    kChunk        = kN / WPB;            // 16384 points per WG
constexpr int    kPtsPerThread = kChunk / THREADS;    // 32
constexpr int    kPieceP       = 2048;                // staging piece (points)
constexpr int    kPieces       = kChunk / kPieceP;    // 8
constexpr int    kPieceBytes   = kPieceP * 12;        // 24576 B (16B aligned)
constexpr int    kB128PerThr   = kPieceBytes / 16 / THREADS;  // 3
constexpr size_t kPtsBytes     = (size_t)kChunk * 16; // 262144 B
constexpr size_t kShmem        = kPtsBytes + (size_t)kPieceBytes; // 286720 B

__device__ __forceinline__ unsigned long long shflxor64(unsigned long long v, int m) {
  unsigned lo = (unsigned)v, hi = (unsigned)(v >> 32);
  lo = __shfl_xor(lo, m, 32);
  hi = __shfl_xor(hi, m, 32);
  return ((unsigned long long)hi << 32) | lo;
}

extern "C" __global__ void fps_init(unsigned long long* slots, unsigned* cnt, int B) {
  int i = threadIdx.x + blockIdx.x * blockDim.x;
  if (i < B * 3) slots[i] = 0ull;
  if (i < B)     cnt[i]   = 0u;
}

extern "C" __global__ __launch_bounds__(THREADS)
void fps_kernel(const float* __restrict__ x, const int* __restrict__ finit,
                int* __restrict__ out, int N, int npoint,
                unsigned long long* slots, unsigned* cnt)
{
  extern __shared__ __align__(16) char smem[];
  float4* pts   = (float4*)smem;                  // [kChunk] x,y,z,|p|^2
  float*  stage = (float*)(smem + kPtsBytes);     // raw AoS staging buffer
  __shared__ unsigned long long wkey[WAVES];
  __shared__ float bc[4];                         // centroid x,y,z,|c|^2
  __shared__ int   bcIdx;

  const int tid     = threadIdx.x;
  const int b       = blockIdx.x / WPB;
  const int wg      = blockIdx.x % WPB;
  const int baseIdx = wg * kChunk;                // batch-relative chunk start
  const float* src  = x + ((size_t)b * N + baseIdx) * 3;

  // ---- Stage AoS xyz chunk -> LDS (async), convert to float4(x,y,z,pp) ----
  const unsigned stageLds = (unsigned)(size_t)(void*)stage;  // low 32b = LDS offset
  for (int p = 0; p < kPieces; ++p) {
    unsigned long long gbase =
        (unsigned long long)(size_t)src + (unsigned long long)p * kPieceBytes;
#pragma unroll
    for (int j = 0; j < kB128PerThr; ++j) {
      unsigned off = (unsigned)(tid + j * THREADS) * 16u;
      unsigned dst = stageLds + off;
      asm volatile("global_load_async_to_lds_b128 %0, %1, %2"
                   :: "v"(dst), "v"(off), "s"(gbase) : "memory");
    }
    asm volatile("s_wait_asynccnt 0" ::: "memory");
    __syncthreads();
#pragma unroll
    for (int j = 0; j < kPieceP / THREADS; ++j) {
      int i = tid + j * THREADS;                  // point within piece
      float px = stage[i * 3 + 0];
      float py = stage[i * 3 + 1];
      float pz = stage[i * 3 + 2];
      pts[p * kPieceP + i] = make_float4(px, py, pz, px * px + py * py + pz * pz);
    }
    __syncthreads();
  }

  // ---- Init running distances (VGPR-resident) and first centroid ----
  float dist[kPtsPerThread];
#pragma unroll
  for (int k = 0; k < kPtsPerThread; ++k) dist[k] = 1.0e10f;

  if (tid == 0) {
    int f = finit[b];
    bcIdx = f;
    const float* c = x + ((size_t)b * N + f) * 3;
    float cx = c[0], cy = c[1], cz = c[2];
    bc[0] = cx; bc[1] = cy; bc[2] = cz; bc[3] = cx * cx + cy * cy + cz * cz;
  }
  __syncthreads();

  volatile unsigned* vcnt = cnt + b;
  unsigned long long* slotBase = slots + (size_t)b * 3;
  const int lane = tid & 31;
  const int wv   = tid >> 5;

  for (int t = 0; t < npoint; ++t) {
    if (wg == 0 && tid == 0) out[(size_t)b * npoint + t] = bcIdx;
    if (t == npoint - 1) break;

    // scan: dist = min(dist, |p|^2 + |c|^2 - 2 p.c), track local argmax
    const float cx = bc[0], cy = bc[1], cz = bc[2], cpp = bc[3];
    float bv = -1.0f; int bk = 0;
#pragma unroll
    for (int k = 0; k < kPtsPerThread; ++k) {
      float4 p  = pts[tid + k * THREADS];
      float dot = fmaf(p.x, cx, fmaf(p.y, cy, p.z * cz));
      float d   = fmaf(-2.0f, dot, p.w) + cpp;
      float nd  = fminf(dist[k], d);
      dist[k]   = nd;
      if (nd > bv) { bv = nd; bk = k; }           // strict > keeps smallest idx
    }
    unsigned gidx = (unsigned)(baseIdx + tid + bk * THREADS);
    unsigned long long key =
        ((unsigned long long)__float_as_uint(bv) << 32) |
        (unsigned long long)(0xFFFFFFFFu - gidx);  // max-value, then min-index

    // wave32 reduction, then per-WG reduction in LDS
#pragma unroll
    for (int m = 16; m >= 1; m >>= 1) {
      unsigned long long o = shflxor64(key, m);
      key = (o > key) ? o : key;
    }
    if (lane == 0) wkey[wv] = key;
    __syncthreads();

    if (tid == 0) {
      unsigned long long k2 = wkey[0];
#pragma unroll
      for (int w = 1; w < WAVES; ++w) {
        unsigned long long o = wkey[w];
        if (o > k2) k2 = o;
      }
      // triple-buffered slot protocol: reset the t+1 buffer (last read at t-2)
      atomicExch(&slotBase[(t + 1) % 3], 0ull);
      atomicMax(&slotBase[t % 3], k2);
      __threadfence();
      atomicAdd(cnt + b, 1u);
      const unsigned target = (unsigned)(WPB * (t + 1));
      while (*vcnt < target) __builtin_amdgcn_s_sleep(2);
      __threadfence();
      unsigned long long win = *(volatile unsigned long long*)&slotBase[t % 3];
      unsigned g = 0xFFFFFFFFu - (unsigned)(win & 0xFFFFFFFFull);
      bcIdx = (int)g;
      const float* c = x + ((size_t)b * N + g) * 3;
      float nx = c[0], ny = c[1], nz = c[2];
      bc[0] = nx; bc[1] = ny; bc[2] = nz; bc[3] = nx * nx + ny * ny + nz * nz;
    }
    __syncthreads();
  }
}

extern "C" void kernel_launch(void* const* d_in, const int* in_sizes, int n_in,
                              void* d_out, int out_size, void* d_ws, size_t ws_size,
                              hipStream_t stream) {
  const float* x     = (const float*)d_in[0];
  const int*   finit = (const int*)d_in[1];
  (void)n_in; (void)ws_size;

  const int B      = in_sizes[1];             // 16
  const int N      = in_sizes[0] / (B * 3);   // 131072
  const int npoint = out_size / B;            // 4096
  int* out = (int*)d_out;

  unsigned long long* slots = (unsigned long long*)d_ws;  // B*3 u64
  unsigned*           cnt   = (unsigned*)(slots + (size_t)B * 3);

  fps_init<<<1, 256, 0, stream>>>(slots, cnt, B);

  (void)hipFuncSetAttribute((const void*)fps_kernel,
                            hipFuncAttributeMaxDynamicSharedMemorySize,
                            (int)kShmem);
  fps_kernel<<<B * WPB, THREADS, kShmem, stream>>>(x, finit, out, N, npoint,
                                                   slots, cnt);
}